// gaussian_divergence_52544629899793
// MI455X (gfx1250) — compile-verified
//
#include <hip/hip_runtime.h>

typedef float v2f __attribute__((ext_vector_type(2)));
typedef float v8f __attribute__((ext_vector_type(8)));

#define Ld 6
#define Nd 2
#define Qd 64
#define Cd 256

// One workgroup per (l,n) pair: 12 workgroups, 256 threads (8 wave32s) each.
// Phase 1: column sums S[c], SS[c] of the 64x256 tile via V_WMMA_F32_16X16X4_F32
//          (A = ones 16x4, B = 4x16 data chunk, f32 accumulate -> exact).
// Phase 2: leave-one-out KL per (q,c), reduced over C with lane shuffles.
__global__ __launch_bounds__(256) void gaussian_divergence_kernel(
    const float* __restrict__ hs, float* __restrict__ out) {
  const int bid  = blockIdx.x;                     // l*N + n, 0..11
  const float* X = hs + (size_t)bid * (Qd * Cd);   // 64 x 256 tile
  const int tid  = threadIdx.x;
  const int lane = tid & 31;
  const int wv   = tid >> 5;      // wave id 0..7
  const int hlf  = lane >> 4;     // 0: lanes 0-15, 1: lanes 16-31
  const int col  = lane & 15;     // column within 16-wide block

  // Pull the 64KB tile toward the caches (gfx1250 global_prefetch path).
  __builtin_prefetch(X + tid * 64, 0, 0);        // 256B stride covers the tile
  __builtin_prefetch(X + tid * 64 + 32, 0, 0);   // fill the in-between lines

  const v2f ones = {1.0f, 1.0f};

  // Per-lane stats for channels c_k = lane + 32k (k = 0..7)
  float Sv[8], Qv[8];

  #pragma unroll
  for (int b = 0; b < 16; ++b) {                 // 16-channel block
    v8f accS = {0.f, 0.f, 0.f, 0.f, 0.f, 0.f, 0.f, 0.f};
    v8f accQ = {0.f, 0.f, 0.f, 0.f, 0.f, 0.f, 0.f, 0.f};
    const float* base = X + b * 16 + col;
    #pragma unroll 4
    for (int k = 0; k < 16; ++k) {               // 4 rows per chunk
      const int r0 = 4 * k + 2 * hlf;            // B VGPR0 row (K=0 / K=2 halves)
      const float x0 = base[(r0 + 0) * Cd];      // K = 2*hlf
      const float x1 = base[(r0 + 1) * Cd];      // K = 2*hlf + 1
      v2f bS = {x0, x1};
      v2f bQ = {x0 * x0, x1 * x1};
      // D = ones(16x4) * B(4x16) + C  => column sums accumulated in f32
      accS = __builtin_amdgcn_wmma_f32_16x16x4_f32(
          false, ones, false, bS, (short)0, accS, false, false);
      accQ = __builtin_amdgcn_wmma_f32_16x16x4_f32(
          false, ones, false, bQ, (short)0, accQ, false, false);
    }
    // Every lane's accS[0] holds S[16*b + col] (all D rows are identical).
    // Lane needs channel c_k = lane + 32k = 16*(2k + hlf) + col  -> block b = 2k + hlf.
    if ((b & 1) == hlf) {
      Sv[b >> 1] = accS[0];
      Qv[b >> 1] = accQ[0];
    }
  }

  // Full-sample mean / var (ddof=1, n=64)
  float meanv[8], varv[8];
  #pragma unroll
  for (int k = 0; k < 8; ++k) {
    const float m = Sv[k] * (1.0f / 64.0f);
    meanv[k] = m;
    varv[k]  = (Qv[k] - Sv[k] * m) * (1.0f / 63.0f);
  }

  // Each wave handles q = 8*wv .. 8*wv+7 ; excluded row r = (q-1) mod 64.
  #pragma unroll
  for (int i = 0; i < 8; ++i) {
    const int q = wv * 8 + i;
    const int r = (q + 63) & 63;
    const float* row = X + r * Cd;
    float part = 0.0f;
    #pragma unroll
    for (int k = 0; k < 8; ++k) {
      const float x  = row[lane + 32 * k];
      const float Sp = Sv[k] - x;              // leave-one-out sum
      const float Qp = Qv[k] - x * x;          // leave-one-out sum of squares
      const float rm_mean = Sp * (1.0f / 63.0f);
      const float rm_var  = (Qp - Sp * rm_mean) * (1.0f / 62.0f); // ddof=1, n=63
      const float sp = varv[k];
      const float dm = meanv[k] - rm_mean;
      const float kl = logf(rm_var / sp)
                     + (sp * sp + dm * dm) / (2.0f * rm_var * rm_var)
                     - 0.5f;
      part += kl;
    }
    // wave32 reduction over lanes (channels)
    #pragma unroll
    for (int off = 16; off >= 1; off >>= 1)
      part += __shfl_xor(part, off, 32);
    if (lane == 0)
      out[bid * Qd + q] = part * (1.0f / (float)Cd);
  }
}

extern "C" void kernel_launch(void* const* d_in, const int* in_sizes, int n_in,
                              void* d_out, int out_size, void* d_ws, size_t ws_size,
                              hipStream_t stream) {
  const float* hs = (const float*)d_in[0];
  float* out = (float*)d_out;
  gaussian_divergence_kernel<<<Ld * Nd, 256, 0, stream>>>(hs, out);
}